// GrowingGeneratorTriplane_62818191671789
// MI455X (gfx1250) — compile-verified
//
#include <hip/hip_runtime.h>
#include <hip/hip_bf16.h>

typedef __attribute__((ext_vector_type(2))) float v2f;
typedef __attribute__((ext_vector_type(8))) float v8f;

#define R 128
#define C 32
#define PLANE_PIX (R * R)   // 16384

// ---------------------------------------------------------------------------
// Stage 1: p[plane][pix] = sum_c feat[plane][c][pix] * w[plane*32 + c]
// One wave -> 16 pixels via 8 chained V_WMMA_F32_16X16X4_F32 (K=4 each, C=32).
//   A[m][k] = w[4s+k]           (broadcast over M; exact f32 WMMA)
//   B[k][n] = feat[4s+k][pix_n] (16 pixels on N)
//   D[m][n] = p[pix_n] for every m  -> lanes 0..15 read acc[0]
// 3 planes * 1024 tiles = 3072 waves = 384 blocks * 8 waves.
// ---------------------------------------------------------------------------
__global__ __launch_bounds__(256) void triplane_reduce_wmma(
    const float* __restrict__ yz, const float* __restrict__ xz,
    const float* __restrict__ xy, const float* __restrict__ W,
    float* __restrict__ pout)
{
    const int lane  = threadIdx.x & 31;
    const int wave  = threadIdx.x >> 5;
    const int tile  = blockIdx.x * 8 + wave;   // 0..3071
    const int plane = tile >> 10;              // 0..2
    const int tp    = tile & 1023;             // tile within plane
    const int pix   = (tp << 4) + (lane & 15); // N = pixel handled by this lane
    const int khalf = (lane >> 4) << 1;        // K sub-offset: 0 (lanes 0-15) / 2 (16-31)

    const float* feat = (plane == 0) ? yz : (plane == 1) ? xz : xy;
    const float* w    = W + plane * C;

    v8f acc = {};
#pragma unroll
    for (int s = 0; s < 8; ++s) {
        const int c0 = s * 4 + khalf;
        v2f a, b;
        a.x = w[c0];
        a.y = w[c0 + 1];
        b.x = feat[(size_t)c0 * PLANE_PIX + pix];
        b.y = feat[(size_t)(c0 + 1) * PLANE_PIX + pix];
        // 8 args: (neg_a, A, neg_b, B, c_mod, C, reuse_a, reuse_b)
        acc = __builtin_amdgcn_wmma_f32_16x16x4_f32(
            false, a, false, b, (short)0, acc, false, false);
    }
    // Every M-row of D equals p[pix_n]; C/D layout: VGPR0, lanes 0-15 -> N=lane.
    if (lane < 16)
        pout[plane * PLANE_PIX + (tp << 4) + lane] = acc[0];
}

// ---------------------------------------------------------------------------
// Stage 2: out[x,y,z] = sigmoid(pyz[y,z] + pxz[x,z] + pxy[x,y] + b)
// One wave per (x,y) row; 32 lanes * float4 over z -> global_store_b128.
// 16384 rows = 2048 blocks * 8 waves. Pure store-bandwidth bound (8 MB).
// ---------------------------------------------------------------------------
__global__ __launch_bounds__(256) void triplane_bcast_sigmoid(
    const float* __restrict__ p,      // [3*16384]: pyz, pxz, pxy
    const float* __restrict__ bptr,
    float* __restrict__ out)
{
    const int lane = threadIdx.x & 31;
    const int wave = threadIdx.x >> 5;
    const int r    = blockIdx.x * 8 + wave;  // r = x*128 + y
    const int x    = r >> 7;
    const int y    = r & 127;

    const float* pyz = p;
    const float* pxz = p + PLANE_PIX;
    const float* pxy = p + 2 * PLANE_PIX;

    const float base = pxy[x * R + y] + bptr[0];
    const int z0 = lane * 4;

    const float4 vyz = *reinterpret_cast<const float4*>(pyz + y * R + z0);
    const float4 vxz = *reinterpret_cast<const float4*>(pxz + x * R + z0);

    float4 o;
    o.x = 1.0f / (1.0f + __expf(-(vyz.x + vxz.x + base)));
    o.y = 1.0f / (1.0f + __expf(-(vyz.y + vxz.y + base)));
    o.z = 1.0f / (1.0f + __expf(-(vyz.z + vxz.z + base)));
    o.w = 1.0f / (1.0f + __expf(-(vyz.w + vxz.w + base)));

    *reinterpret_cast<float4*>(out + ((size_t)r << 7) + z0) = o;
}

extern "C" void kernel_launch(void* const* d_in, const int* in_sizes, int n_in,
                              void* d_out, int out_size, void* d_ws, size_t ws_size,
                              hipStream_t stream) {
    const float* yz = (const float*)d_in[0];
    const float* xz = (const float*)d_in[1];
    const float* xy = (const float*)d_in[2];
    const float* W  = (const float*)d_in[3];
    const float* b  = (const float*)d_in[4];
    float* out = (float*)d_out;
    float* pws = (float*)d_ws;   // needs 3*16384*4 = 196608 bytes of scratch

    // Stage 1: WMMA channel reduction into L2-resident workspace (192 KB).
    triplane_reduce_wmma<<<384, 256, 0, stream>>>(yz, xz, xy, W, pws);
    // Stage 2: broadcast-add + sigmoid, 8 MB of coalesced b128 stores.
    triplane_bcast_sigmoid<<<2048, 256, 0, stream>>>(pws, b, out);
}